// GraphAttentionLayer_15685220565556
// MI455X (gfx1250) — compile-verified
//
#include <hip/hip_runtime.h>
#include <hip/hip_bf16.h>
#include <math.h>

#define N_NODES 100000
#define N_EDGES 1600000
#define IN_DIM 256
#define OUT_DIM 128
#define HEADS 4
#define HEAD_DIM 32
#define NEG_SLOPE 0.2f

typedef __attribute__((ext_vector_type(2))) float v2f;
typedef __attribute__((ext_vector_type(8))) float v8f;

// ---------- atomics ----------
__device__ __forceinline__ void atomAddF(float* p, float v) {
  unsafeAtomicAdd(p, v);   // global_atomic_add_f32 on CDNA
}
// Exact float atomic max via sign-aware integer atomics (init must be -inf).
__device__ __forceinline__ void atomMaxF(float* addr, float val) {
  if (val >= 0.0f) atomicMax((int*)addr, __float_as_int(val));
  else             atomicMin((unsigned int*)addr, __float_as_uint(val));
}

// ---------- 1) h = x @ W^T + b via V_WMMA_F32_16X16X4_F32 ----------
// One wave computes a 16(M) x 128(N) strip: 8 f32 accumulators, K-loop of 64 WMMAs x 8 tiles.
__global__ __launch_bounds__(256) void gat_gemm_wmma(
    const float* __restrict__ x, const float* __restrict__ W,
    const float* __restrict__ bias, float* __restrict__ h) {
  const int wave = threadIdx.x >> 5;
  const int lane = threadIdx.x & 31;
  const int mtile = blockIdx.x * 8 + wave;
  if (mtile >= N_NODES / 16) return;          // wave-uniform: EXEC stays all-1s for WMMA

  const int col   = lane & 15;                // N index inside 16-wide tile / M row for A
  const int khalf = lane >> 4;                // 0: K={0,1}, 1: K={2,3} (A/B fragment layout)
  const int row   = mtile * 16 + col;         // A-matrix row for this lane

  v8f acc[8];
  #pragma unroll
  for (int t = 0; t < 8; ++t) {               // init accumulators with bias (broadcast per column)
    const float bv = bias[t * 16 + col];
    acc[t] = (v8f){bv, bv, bv, bv, bv, bv, bv, bv};
  }

  const float* __restrict__ xrow = x + (size_t)row * IN_DIM;
  for (int k = 0; k < IN_DIM; k += 4) {
    const int kk = k + khalf * 2;
    const v2f a = *(const v2f*)(xrow + kk);   // A[16x4] fragment: 2 consecutive K values
    #pragma unroll
    for (int t = 0; t < 8; ++t) {
      // B[4x16] fragment: B[kk][n] = W[n][kk]  (h = x @ W^T)
      const v2f b = *(const v2f*)(W + (size_t)(t * 16 + col) * IN_DIM + kk);
      acc[t] = __builtin_amdgcn_wmma_f32_16x16x4_f32(
          /*neg_a=*/false, a, /*neg_b=*/false, b,
          /*c_mod=*/(short)0, acc[t], /*reuse_a=*/false, /*reuse_b=*/false);
    }
  }

  // C/D layout: VGPR r -> row (lane/16)*8 + r, col = lane%16
  const int rbase = mtile * 16 + khalf * 8;
  #pragma unroll
  for (int t = 0; t < 8; ++t) {
    const int c0 = t * 16 + col;
    #pragma unroll
    for (int r = 0; r < 8; ++r)
      h[(size_t)(rbase + r) * OUT_DIM + c0] = acc[t][r];
  }
}

// ---------- 2) per-(node,head) attention scores ----------
__global__ __launch_bounds__(256) void gat_scores(
    const float* __restrict__ h, const float* __restrict__ att,
    float* __restrict__ s_src, float* __restrict__ s_tgt) {
  const int i = blockIdx.x * blockDim.x + threadIdx.x;   // node*HEADS + head
  if (i >= N_NODES * HEADS) return;
  const int head = i & (HEADS - 1);
  const float* __restrict__ hp = h + (size_t)(i >> 2) * OUT_DIM + head * HEAD_DIM;
  float ss = 0.f, st = 0.f;
  #pragma unroll
  for (int d = 0; d < HEAD_DIM; ++d) {
    const float v = hp[d];
    ss += v * att[d];
    st += v * att[HEAD_DIM + d];
  }
  s_src[i] = ss;
  s_tgt[i] = st;
}

// ---------- 3) init seg_max=-inf, denom=0, out=0 ----------
__global__ __launch_bounds__(256) void gat_init(
    float* __restrict__ seg_max, float* __restrict__ denom, float* __restrict__ out) {
  const int i = blockIdx.x * blockDim.x + threadIdx.x;
  if (i < N_NODES * HEADS) { seg_max[i] = -INFINITY; denom[i] = 0.0f; }
  if (i < N_NODES * OUT_DIM) out[i] = 0.0f;
}

// ---------- 4) per-edge leaky-relu score -> segment max ----------
__global__ __launch_bounds__(256) void gat_edge_max(
    const long long* __restrict__ src, const long long* __restrict__ tgt,
    const float* __restrict__ s_src, const float* __restrict__ s_tgt,
    float* __restrict__ seg_max) {
  const int i = blockIdx.x * blockDim.x + threadIdx.x;   // edge*HEADS + head
  if (i >= N_EDGES * HEADS) return;
  const int e = i >> 2, hd = i & 3;
  const long long s = src[e], t = tgt[e];
  float v = s_src[s * HEADS + hd] + s_tgt[t * HEADS + hd];
  v = (v >= 0.0f) ? v : NEG_SLOPE * v;
  atomMaxF(&seg_max[t * HEADS + hd], v);
}

// ---------- 5) per-edge exp(e - max) -> segment sum (denominator) ----------
__global__ __launch_bounds__(256) void gat_edge_sum(
    const long long* __restrict__ src, const long long* __restrict__ tgt,
    const float* __restrict__ s_src, const float* __restrict__ s_tgt,
    const float* __restrict__ seg_max, float* __restrict__ denom) {
  const int i = blockIdx.x * blockDim.x + threadIdx.x;
  if (i >= N_EDGES * HEADS) return;
  const int e = i >> 2, hd = i & 3;
  const long long s = src[e], t = tgt[e];
  float v = s_src[s * HEADS + hd] + s_tgt[t * HEADS + hd];
  v = (v >= 0.0f) ? v : NEG_SLOPE * v;
  atomAddF(&denom[t * HEADS + hd], expf(v - seg_max[t * HEADS + hd]));
}

// ---------- 6) scatter: out[tgt] += h[src] * alpha ----------
// One thread per (edge, out-channel). Lanes of a wave share edge+head -> the 4
// scalar gathers broadcast from L2; h (51MB) and all score arrays are L2-resident.
__global__ __launch_bounds__(256) void gat_scatter(
    const long long* __restrict__ src, const long long* __restrict__ tgt,
    const float* __restrict__ s_src, const float* __restrict__ s_tgt,
    const float* __restrict__ seg_max, const float* __restrict__ denom,
    const float* __restrict__ h, float* __restrict__ out) {
  const long long idx = (long long)blockIdx.x * blockDim.x + threadIdx.x;
  if (idx >= (long long)N_EDGES * OUT_DIM) return;
  const int e = (int)(idx >> 7);
  const int c = (int)(idx & 127);            // head*32 + d
  const int hd = c >> 5;
  const long long s = src[e], t = tgt[e];
  float v = s_src[s * HEADS + hd] + s_tgt[t * HEADS + hd];
  v = (v >= 0.0f) ? v : NEG_SLOPE * v;
  const float alpha = expf(v - seg_max[t * HEADS + hd]) / denom[t * HEADS + hd];
  atomAddF(&out[(size_t)t * OUT_DIM + c], h[(size_t)s * OUT_DIM + c] * alpha);
}

extern "C" void kernel_launch(void* const* d_in, const int* in_sizes, int n_in,
                              void* d_out, int out_size, void* d_ws, size_t ws_size,
                              hipStream_t stream) {
  const float*     x    = (const float*)d_in[0];
  const long long* ei   = (const long long*)d_in[1];   // int64, shape (2, N_EDGES)
  const float*     W    = (const float*)d_in[2];       // (128, 256) row-major
  const float*     bias = (const float*)d_in[3];       // (128,)
  const float*     att  = (const float*)d_in[4];       // (64,)
  float*           out  = (float*)d_out;

  const long long* src = ei;
  const long long* tgt = ei + N_EDGES;

  // workspace layout (floats): h | s_src | s_tgt | seg_max | denom  (~57.6 MB)
  float* h       = (float*)d_ws;
  float* s_src   = h + (size_t)N_NODES * OUT_DIM;
  float* s_tgt   = s_src + (size_t)N_NODES * HEADS;
  float* seg_max = s_tgt + (size_t)N_NODES * HEADS;
  float* denom   = seg_max + (size_t)N_NODES * HEADS;

  // 1) GEMM: 6250 m-tiles, 8 waves/block
  gat_gemm_wmma<<<(N_NODES / 16 + 7) / 8, 256, 0, stream>>>(x, W, bias, h);
  // 2) scores
  gat_scores<<<(N_NODES * HEADS + 255) / 256, 256, 0, stream>>>(h, att, s_src, s_tgt);
  // 3) init
  gat_init<<<(N_NODES * OUT_DIM + 255) / 256, 256, 0, stream>>>(seg_max, denom, out);
  // 4) segment max
  gat_edge_max<<<(N_EDGES * HEADS + 255) / 256, 256, 0, stream>>>(src, tgt, s_src, s_tgt, seg_max);
  // 5) softmax denominator
  gat_edge_sum<<<(N_EDGES * HEADS + 255) / 256, 256, 0, stream>>>(src, tgt, s_src, s_tgt, seg_max, denom);
  // 6) weighted scatter
  const long long total = (long long)N_EDGES * OUT_DIM;
  gat_scatter<<<(unsigned)((total + 255) / 256), 256, 0, stream>>>(src, tgt, s_src, s_tgt, seg_max, denom, h, out);
}